// SRLEmbeddings_1589137899520
// MI455X (gfx1250) — compile-verified
//
#include <hip/hip_runtime.h>

typedef float v2f __attribute__((ext_vector_type(2)));
typedef float v8f __attribute__((ext_vector_type(8)));

#define Bn 16
#define Sn 32
#define Ln 128
#define Dn 768
#define An 6
#define Mn 10
#define VOCABn 200
#define WS 33   // K-major LDS stride for W (dodges 64-bank conflicts)

__global__ __launch_bounds__(256)
void srl_wmma_kernel(const int* __restrict__ sids,
                     const int* __restrict__ mask,
                     const float* __restrict__ emb,
                     const int* __restrict__ pred,
                     const int* __restrict__ arg0,
                     const int* __restrict__ arg1,
                     float* __restrict__ out)
{
    __shared__ float Wl[Ln * WS];       // weight matrix, stored [K=128][M=32(+pad)]
    __shared__ int   sid_sh[Ln];
    __shared__ int   msk_sh[Ln];
    __shared__ int   hist[VOCABn];      // token histogram over this sentence
    __shared__ int   mcnt;              // attention-mask count
    __shared__ int   tok_sh[19];        // winning token per row (rows 1..18)
    __shared__ float inv_sh[19];        // 1/count per row

    const int bs   = blockIdx.x;        // b*S + s, 0..511
    const int t    = threadIdx.x;       // 0..255
    const int lane = t & 31;            // wave32
    const int wave = t >> 5;            // 0..7

    // ---- Phase 0: stage ids + mask, zero histogram ----
    if (t < VOCABn) hist[t] = 0;
    if (t == 0)     mcnt = 0;
    int mySid = 0, myMsk = 0;
    if (t < Ln)          { mySid = sids[bs * Ln + t];        sid_sh[t] = mySid; }
    else if (t < 2 * Ln) { myMsk = mask[bs * Ln + (t - Ln)]; msk_sh[t - Ln] = myMsk; }
    __syncthreads();

    // ---- Phase 1a: histogram + mask count via LDS atomics (parallel, no L-scans) ----
    if (t < Ln)               atomicAdd(&hist[mySid], 1);
    else if (t < 2 * Ln) { if (myMsk) atomicAdd(&mcnt, 1); }
    __syncthreads();

    // ---- Phase 1b: per-row token selection (last arg token with >=1 match wins) ----
    if (t < 19) {
        if (t == 0) {
            const int c = mcnt;
            tok_sh[0] = -1;
            inv_sh[0] = 1.0f / (float)(c > 0 ? c : 1);
        } else {
            const int r = t;
            int a; const int* ids;
            if (r <= 6)       { a = r - 1;  ids = pred; }
            else if (r <= 12) { a = r - 7;  ids = arg0; }
            else              { a = r - 13; ids = arg1; }
            ids += (bs * An + a) * Mn;
            int tok = 0, c = 0;
            #pragma unroll
            for (int m = 0; m < Mn; ++m) {
                const int tm = ids[m];
                if (tm != 0) {
                    const int cm = hist[tm];
                    if (cm > 0) { tok = tm; c = cm; }
                }
            }
            tok_sh[r] = tok;                       // tok==0 -> no match (inv forces row to 0)
            inv_sh[r] = (c > 0) ? 1.0f / (float)c : 0.0f;
        }
    }
    __syncthreads();

    // ---- Phase 1c: fill W[32 x 128] cooperatively (16 entries / thread, conflict-free) ----
    #pragma unroll
    for (int i = 0; i < 16; ++i) {
        const int idx = t + i * 256;               // 0..4095 over (l,r)
        const int l = idx >> 5;
        const int r = idx & 31;
        float w;
        if (r == 0)       w = msk_sh[l] ? inv_sh[0] : 0.0f;
        else if (r <= 18) w = (sid_sh[l] == tok_sh[r]) ? inv_sh[r] : 0.0f;
        else              w = 0.0f;                // pad rows 19..31
        Wl[l * WS + r] = w;
    }
    __syncthreads();

    // ---- Phase 2: O[32x768] = W[32x128] * E[128x768] via v_wmma_f32_16x16x4_f32 ----
    v8f acc[2][6];
    #pragma unroll
    for (int mt = 0; mt < 2; ++mt)
        #pragma unroll
        for (int j = 0; j < 6; ++j) {
            v8f z = {0.f, 0.f, 0.f, 0.f, 0.f, 0.f, 0.f, 0.f};
            acc[mt][j] = z;
        }

    const int lm = lane & 15;
    const int kk = (lane >> 4) << 1;                 // lanes 0-15 -> K {0,1}; 16-31 -> K {2,3}
    const float* Ebs = emb + (size_t)bs * Ln * Dn;
    const int nb = wave * 96 + lm;                   // this wave's column base + lane

    for (int kc = 0; kc < 32; ++kc) {
        const int k0 = (kc << 2) + kk;               // this lane's two K rows: k0, k0+1
        v2f a0, a1;
        a0.x = Wl[ k0      * WS + lm];
        a0.y = Wl[(k0 + 1) * WS + lm];
        a1.x = Wl[ k0      * WS + 16 + lm];
        a1.y = Wl[(k0 + 1) * WS + 16 + lm];
        const float* Er = Ebs + (size_t)k0 * Dn;
        #pragma unroll
        for (int j = 0; j < 6; ++j) {
            v2f b;
            b.x = __builtin_nontemporal_load(&Er[     nb + j * 16]);   // row k0   (single-use stream)
            b.y = __builtin_nontemporal_load(&Er[Dn + nb + j * 16]);   // row k0+1
            acc[0][j] = __builtin_amdgcn_wmma_f32_16x16x4_f32(
                false, a0, false, b, (short)0, acc[0][j], false, false);
            acc[1][j] = __builtin_amdgcn_wmma_f32_16x16x4_f32(
                false, a1, false, b, (short)0, acc[1][j], false, false);
        }
    }

    // ---- Phase 3: scatter D rows to the four concatenated outputs ----
    const size_t O1 = (size_t)Bn * Sn * Dn;
    const size_t O2 = O1 + (size_t)Bn * Sn * An * Dn;
    const size_t O3 = O2 + (size_t)Bn * Sn * An * Dn;
    const int mhalf = (lane >> 4) << 3;              // VGPR r holds M=r (lo half) / r+8 (hi half)

    #pragma unroll
    for (int mt = 0; mt < 2; ++mt) {
        #pragma unroll
        for (int j = 0; j < 6; ++j) {
            const int n = nb + j * 16;
            #pragma unroll
            for (int r = 0; r < 8; ++r) {
                const int m = mt * 16 + r + mhalf;
                const float val = acc[mt][j][r];
                if (m == 0)
                    __builtin_nontemporal_store(val, &out[(size_t)bs * Dn + n]);
                else if (m <= 6)
                    __builtin_nontemporal_store(val, &out[O1 + ((size_t)bs * An + (m - 1))  * Dn + n]);
                else if (m <= 12)
                    __builtin_nontemporal_store(val, &out[O2 + ((size_t)bs * An + (m - 7))  * Dn + n]);
                else if (m <= 18)
                    __builtin_nontemporal_store(val, &out[O3 + ((size_t)bs * An + (m - 13)) * Dn + n]);
                // m >= 19: padding rows, dropped
            }
        }
    }
}

extern "C" void kernel_launch(void* const* d_in, const int* in_sizes, int n_in,
                              void* d_out, int out_size, void* d_ws, size_t ws_size,
                              hipStream_t stream)
{
    (void)in_sizes; (void)n_in; (void)out_size; (void)d_ws; (void)ws_size;
    const int*   sids = (const int*)  d_in[0];
    const int*   mask = (const int*)  d_in[1];
    const float* emb  = (const float*)d_in[2];
    const int*   pred = (const int*)  d_in[3];
    const int*   a0   = (const int*)  d_in[4];
    const int*   a1   = (const int*)  d_in[5];
    float* out = (float*)d_out;

    dim3 grid(Bn * Sn);   // 512 workgroups, one per (b,s)
    dim3 block(256);      // 8 wave32
    hipLaunchKernelGGL(srl_wmma_kernel, grid, block, 0, stream,
                       sids, mask, emb, pred, a0, a1, out);
}